// HematoxylinFFTModel_86234353369343
// MI455X (gfx1250) — compile-verified
//
#include <hip/hip_runtime.h>
#include <math.h>

// ---------------------------------------------------------------------------
// HematoxylinFFTModel for MI455X (gfx1250, wave32).
// Four-step 512-pt FFT (512 = 16 x 32) on V_WMMA_F32_16X16X4_F32, with
// block-level LDS transpose so all global stores are 128B-coalesced.
// ---------------------------------------------------------------------------

typedef float v2f __attribute__((ext_vector_type(2)));
typedef float v8f __attribute__((ext_vector_type(8)));

#define NB        64
#define NSZ       512
#define IMG_ELEMS (NSZ * NSZ)            // 262144
#define PLANE     (NB * IMG_ELEMS)       // 16777216 floats per complex plane
#define TAB_SZ    3584                   // w32c|w32s|w16c|w16s|t5c|t5s
#define NPART     1024                   // phase-2 blocks (16 per image)
#define YPITCH    513                    // padded pitch: bank-conflict-free transpose

// hed_from_rgb column 0 (inv of skimage rgb_from_hed), and 1/log(1e-6)
#define HW0  ( 1.8779837f)
#define HW1  (-0.0659081f)
#define HW2  (-0.6019079f)
#define INV_LOG_ADJ (-0.072382415f)
#define FWD_SCALE   (3.814697265625e-6f) // 1/(512*512), fft norm='forward'
#define TWO_PI 6.28318530717958647692f

// D = A(16x4) * B(4x16) + C, f32 WMMA.
__device__ __forceinline__ v8f wmma_f32(v2f a, v2f b, v8f c) {
  return __builtin_amdgcn_wmma_f32_16x16x4_f32(false, a, false, b, (short)0, c,
                                               false, false);
}

// ---------------------------------------------------------------------------
// Tables (w32: 32x32, w16: 16x16, t5: 16x32), row-major [k*N + n],
// angle = -2*pi*k*n/N (numpy forward sign). W16/W32 are symmetric.
// ---------------------------------------------------------------------------
__global__ void k_init(float* __restrict__ tab) {
  int i = blockIdx.x * 256 + threadIdx.x;  // grid covers exactly 1792
  if (i < 1024) {                           // W32
    int k = i >> 5, n = i & 31;
    int m = (k * n) & 31;
    float s, c;
    __sincosf(-(TWO_PI / 32.0f) * (float)m, &s, &c);
    tab[i] = c;  tab[1024 + i] = s;
  } else if (i < 1280) {                    // W16
    int j = i - 1024;
    int k = j >> 4, n = j & 15;
    int m = (k * n) & 15;
    float s, c;
    __sincosf(-(TWO_PI / 16.0f) * (float)m, &s, &c);
    tab[2048 + j] = c;  tab[2304 + j] = s;
  } else if (i < 1792) {                    // T512: omega_512^{n1*k2}
    int j = i - 1280;
    int n1 = j >> 5, k2 = j & 31;
    float s, c;
    __sincosf(-(TWO_PI / 512.0f) * (float)(n1 * k2), &s, &c);
    tab[2560 + j] = c;  tab[3072 + j] = s;
  }
}

extern __shared__ float smem[];

// ---------------------------------------------------------------------------
// Phase 1: 1024-thread block = 32 waves = 32 consecutive image rows.
// Per wave: fused hematoxylin OD + four-step 512-pt real FFT (WMMA).
//   d[n], n = n1 + 16*n2;  Y[n1][k2] = sum_n2 d * W32[n2][k2];
//   Y' = Y .* T512;  Z[k1][k2] = sum_n1 W16[k1][n1] * Y';  k = k2 + 32*k1.
// Z is written in place over the Y' LDS buffer (chunk c only reads/writes
// columns c*16..c*16+15), then the block stores a 32-row tile of the
// TRANSPOSED planes FrT[b][k][r], FiT[b][k][r] with 128B-coalesced writes.
// ---------------------------------------------------------------------------
__global__ __launch_bounds__(1024) void k_phase1(const float* __restrict__ x,
                                                 float* __restrict__ FrT,
                                                 float* __restrict__ FiT,
                                                 const float* __restrict__ gtab) {
  float* s_tab = smem;                         // 3584
  float* s_ybr = smem + TAB_SZ;                // 32*513
  float* s_ybi = s_ybr + 32 * YPITCH;          // 32*513
  const int tid = threadIdx.x;
  for (int i = tid; i < TAB_SZ; i += 1024) s_tab[i] = gtab[i];
  __syncthreads();
  const float* w32c = s_tab;
  const float* w32s = s_tab + 1024;
  const float* w16c = s_tab + 2048;
  const float* w16s = s_tab + 2304;
  const float* t5c  = s_tab + 2560;
  const float* t5s  = s_tab + 3072;

  const int warp = tid >> 5;
  const int lane = tid & 31;
  const int half = lane >> 4;
  const int lm   = lane & 15;
  const int b    = blockIdx.x >> 4;
  const int r0   = (blockIdx.x & 15) << 5;
  const int r    = r0 + warp;

  // ---- load row, fuse stain deconvolution, place into A-operand layout ----
  const size_t xbase = ((size_t)b * 3 * NSZ + (size_t)r) * NSZ;
  const size_t cstr  = (size_t)NSZ * NSZ;
  v2f a[8];
  #pragma unroll
  for (int s = 0; s < 8; ++s) {
    #pragma unroll
    for (int j = 0; j < 2; ++j) {
      int n = lm + 16 * (4 * s + 2 * half + j);
      float rr = fminf(fmaxf(x[xbase + n],            1e-6f), 1.0f);
      float gg = fminf(fmaxf(x[xbase + cstr + n],     1e-6f), 1.0f);
      float bb = fminf(fmaxf(x[xbase + 2 * cstr + n], 1e-6f), 1.0f);
      float od = (logf(rr) * HW0 + logf(gg) * HW1 + logf(bb) * HW2) * INV_LOG_ADJ;
      a[s][j] = fmaxf(od, 0.0f);
    }
  }

  float* ybr = s_ybr + warp * YPITCH;
  float* ybi = s_ybi + warp * YPITCH;

  // ---- step 1: Y = D * W32 (real input); step 2: twiddle, stash in LDS ----
  #pragma unroll
  for (int nc = 0; nc < 2; ++nc) {
    const int ncol = nc * 16 + lm;
    v8f yr = {0.f, 0.f, 0.f, 0.f, 0.f, 0.f, 0.f, 0.f};
    v8f yi = {0.f, 0.f, 0.f, 0.f, 0.f, 0.f, 0.f, 0.f};
    #pragma unroll
    for (int s = 0; s < 8; ++s) {
      const int k0 = 4 * s + 2 * half;
      v2f bc, bs;
      bc[0] = w32c[(k0 + 0) * 32 + ncol]; bc[1] = w32c[(k0 + 1) * 32 + ncol];
      bs[0] = w32s[(k0 + 0) * 32 + ncol]; bs[1] = w32s[(k0 + 1) * 32 + ncol];
      yr = wmma_f32(a[s], bc, yr);
      yi = wmma_f32(a[s], bs, yi);
    }
    #pragma unroll
    for (int j = 0; j < 8; ++j) {
      const int n1 = j + 8 * half;
      const float tc = t5c[n1 * 32 + ncol];
      const float ts = t5s[n1 * 32 + ncol];
      ybr[n1 * 32 + ncol] = yr[j] * tc - yi[j] * ts;
      ybi[n1 * 32 + ncol] = yr[j] * ts + yi[j] * tc;
    }
  }

  // ---- step 3: Z = W16 * Y' (complex x complex), Z -> ybuf in place -------
  #pragma unroll
  for (int nc = 0; nc < 2; ++nc) {
    const int ncol = nc * 16 + lm;
    v8f zr = {0.f, 0.f, 0.f, 0.f, 0.f, 0.f, 0.f, 0.f};
    v8f zi = {0.f, 0.f, 0.f, 0.f, 0.f, 0.f, 0.f, 0.f};
    #pragma unroll
    for (int s = 0; s < 4; ++s) {
      const int k0 = 4 * s + 2 * half;
      v2f ac, asn, nas, br, bi;
      ac[0]  = w16c[(k0 + 0) * 16 + lm];  ac[1]  = w16c[(k0 + 1) * 16 + lm];
      asn[0] = w16s[(k0 + 0) * 16 + lm];  asn[1] = w16s[(k0 + 1) * 16 + lm];
      nas[0] = -asn[0];                   nas[1] = -asn[1];
      br[0] = ybr[(k0 + 0) * 32 + ncol];  br[1] = ybr[(k0 + 1) * 32 + ncol];
      bi[0] = ybi[(k0 + 0) * 32 + ncol];  bi[1] = ybi[(k0 + 1) * 32 + ncol];
      zr = wmma_f32(ac, br, zr);  zr = wmma_f32(nas, bi, zr);
      zi = wmma_f32(ac, bi, zi);  zi = wmma_f32(asn, br, zi);
    }
    #pragma unroll
    for (int j = 0; j < 8; ++j) {
      const int k = ncol + 32 * (j + 8 * half);   // k mod 32 == ncol: no alias
      ybr[k] = zr[j];
      ybi[k] = zi[j];
    }
  }
  __syncthreads();

  // ---- block-wide coalesced transposed store (32 consecutive r per k) -----
  const size_t obase = (size_t)b * IMG_ELEMS;
  #pragma unroll
  for (int it = 0; it < 16; ++it) {
    int idx = it * 1024 + tid;         // 0..16383
    int k  = idx >> 5;
    int rr = idx & 31;
    size_t o = obase + (size_t)k * NSZ + (r0 + rr);
    FrT[o] = s_ybr[rr * YPITCH + k];
    FiT[o] = s_ybi[rr * YPITCH + k];
  }
}

// ---------------------------------------------------------------------------
// Phase 2: 1024-thread block = 32 rows of the transposed planes (32 column
// FFTs). Complex FFT, fused fftshift + |.|/N^2 + log1p staged in LDS, then
// coalesced store to d_out; deterministic block tree reduction of stats.
// ---------------------------------------------------------------------------
__device__ __forceinline__ float blk_red(float v, int tid, float* buf, int op) {
  buf[tid] = v;
  __syncthreads();
  #pragma unroll
  for (int s = 512; s > 0; s >>= 1) {
    if (tid < s) {
      float a0 = buf[tid], b0 = buf[tid + s];
      buf[tid] = (op == 0) ? fminf(a0, b0) : (op == 1) ? fmaxf(a0, b0) : (a0 + b0);
    }
    __syncthreads();
  }
  float r = buf[0];
  __syncthreads();
  return r;
}

__global__ __launch_bounds__(1024) void k_phase2(const float* __restrict__ FrT,
                                                 const float* __restrict__ FiT,
                                                 const float* __restrict__ gtab,
                                                 float* __restrict__ out,
                                                 float* __restrict__ pmin,
                                                 float* __restrict__ pmax,
                                                 float* __restrict__ psum,
                                                 float* __restrict__ psumsq) {
  float* s_tab = smem;                         // 3584
  float* s_ybr = smem + TAB_SZ;                // 32*513 (re-used for log-mag L)
  float* s_ybi = s_ybr + 32 * YPITCH;          // 32*513
  float* s_red = s_ybi + 32 * YPITCH;          // 1024
  const int tid = threadIdx.x;
  for (int i = tid; i < TAB_SZ; i += 1024) s_tab[i] = gtab[i];
  __syncthreads();
  const float* w32c = s_tab;
  const float* w32s = s_tab + 1024;
  const float* w16c = s_tab + 2048;
  const float* w16s = s_tab + 2304;
  const float* t5c  = s_tab + 2560;
  const float* t5s  = s_tab + 3072;

  const int warp = tid >> 5;
  const int lane = tid & 31;
  const int half = lane >> 4;
  const int lm   = lane & 15;
  const int b    = blockIdx.x >> 4;
  const int u0   = (blockIdx.x & 15) << 5;
  const int u    = u0 + warp;                  // horizontal frequency

  const size_t ibase = (size_t)b * IMG_ELEMS + (size_t)u * NSZ;
  v2f ar[8], ai[8];
  #pragma unroll
  for (int s = 0; s < 8; ++s) {
    #pragma unroll
    for (int j = 0; j < 2; ++j) {
      int n = lm + 16 * (4 * s + 2 * half + j);
      ar[s][j] = FrT[ibase + n];
      ai[s][j] = FiT[ibase + n];
    }
  }

  float* ybr = s_ybr + warp * YPITCH;
  float* ybi = s_ybi + warp * YPITCH;

  #pragma unroll
  for (int nc = 0; nc < 2; ++nc) {
    const int ncol = nc * 16 + lm;
    v8f yr = {0.f, 0.f, 0.f, 0.f, 0.f, 0.f, 0.f, 0.f};
    v8f yi = {0.f, 0.f, 0.f, 0.f, 0.f, 0.f, 0.f, 0.f};
    #pragma unroll
    for (int s = 0; s < 8; ++s) {
      const int k0 = 4 * s + 2 * half;
      v2f bc, bs, nas;
      bc[0] = w32c[(k0 + 0) * 32 + ncol]; bc[1] = w32c[(k0 + 1) * 32 + ncol];
      bs[0] = w32s[(k0 + 0) * 32 + ncol]; bs[1] = w32s[(k0 + 1) * 32 + ncol];
      nas[0] = -ai[s][0];                 nas[1] = -ai[s][1];
      yr = wmma_f32(ar[s], bc, yr);  yr = wmma_f32(nas,   bs, yr);
      yi = wmma_f32(ar[s], bs, yi);  yi = wmma_f32(ai[s], bc, yi);
    }
    #pragma unroll
    for (int j = 0; j < 8; ++j) {
      const int n1 = j + 8 * half;
      const float tc = t5c[n1 * 32 + ncol];
      const float ts = t5s[n1 * 32 + ncol];
      ybr[n1 * 32 + ncol] = yr[j] * tc - yi[j] * ts;
      ybi[n1 * 32 + ncol] = yr[j] * ts + yi[j] * tc;
    }
  }

  float lmin = INFINITY, lmax = 0.0f, lsum = 0.0f, lsq = 0.0f;
  #pragma unroll
  for (int nc = 0; nc < 2; ++nc) {
    const int ncol = nc * 16 + lm;
    v8f zr = {0.f, 0.f, 0.f, 0.f, 0.f, 0.f, 0.f, 0.f};
    v8f zi = {0.f, 0.f, 0.f, 0.f, 0.f, 0.f, 0.f, 0.f};
    #pragma unroll
    for (int s = 0; s < 4; ++s) {
      const int k0 = 4 * s + 2 * half;
      v2f ac, asn, nas, br, bi;
      ac[0]  = w16c[(k0 + 0) * 16 + lm];  ac[1]  = w16c[(k0 + 1) * 16 + lm];
      asn[0] = w16s[(k0 + 0) * 16 + lm];  asn[1] = w16s[(k0 + 1) * 16 + lm];
      nas[0] = -asn[0];                   nas[1] = -asn[1];
      br[0] = ybr[(k0 + 0) * 32 + ncol];  br[1] = ybr[(k0 + 1) * 32 + ncol];
      bi[0] = ybi[(k0 + 0) * 32 + ncol];  bi[1] = ybi[(k0 + 1) * 32 + ncol];
      zr = wmma_f32(ac, br, zr);  zr = wmma_f32(nas, bi, zr);
      zi = wmma_f32(ac, bi, zi);  zi = wmma_f32(asn, br, zi);
    }
    #pragma unroll
    for (int j = 0; j < 8; ++j) {
      const int v  = ncol + 32 * (j + 8 * half);  // vertical frequency
      const int vs = (v + 256) & 511;             // fftshift; vs mod 32 == ncol
      float mag = sqrtf(zr[j] * zr[j] + zi[j] * zi[j]) * FWD_SCALE;
      float L   = log1pf(mag);
      ybr[vs] = L;                                // stage shifted log-mag
      lmin = fminf(lmin, L);
      lmax = fmaxf(lmax, L);
      lsum += L;
      lsq  += L * L;
    }
  }

  // deterministic block-level reductions (barriers also fence the L staging)
  float bmin = blk_red(lmin, tid, s_red, 0);
  float bmax = blk_red(lmax, tid, s_red, 1);
  float bsum = blk_red(lsum, tid, s_red, 2);
  float bsq  = blk_red(lsq,  tid, s_red, 2);
  if (tid == 0) {
    pmin[blockIdx.x]   = bmin;
    pmax[blockIdx.x]   = bmax;
    psum[blockIdx.x]   = bsum;
    psumsq[blockIdx.x] = bsq;
  }

  // ---- block-wide coalesced transposed store of log-mag -------------------
  const int us0 = (u0 + 256) & 511;              // fftshift (horizontal)
  const size_t obase = (size_t)b * IMG_ELEMS;
  #pragma unroll
  for (int it = 0; it < 16; ++it) {
    int idx = it * 1024 + tid;
    int vs = idx >> 5;
    int w  = idx & 31;
    out[obase + (size_t)vs * NSZ + (us0 + w)] = s_ybr[w * YPITCH + vs];
  }
}

// ---------------------------------------------------------------------------
// Per-image reduction of the 16 block partials (deterministic tree).
// ---------------------------------------------------------------------------
__global__ __launch_bounds__(16) void k_reduce(const float* __restrict__ pmin,
                                               const float* __restrict__ pmax,
                                               const float* __restrict__ psum,
                                               const float* __restrict__ psumsq,
                                               float* __restrict__ gstat) {
  __shared__ float buf[16];
  const int b = blockIdx.x, t = threadIdx.x;
  #pragma unroll
  for (int w = 0; w < 4; ++w) {
    const float* src = (w == 0) ? pmin : (w == 1) ? pmax : (w == 2) ? psum : psumsq;
    buf[t] = src[b * 16 + t];
    __syncthreads();
    #pragma unroll
    for (int s = 8; s > 0; s >>= 1) {
      if (t < s) {
        float a0 = buf[t], b0 = buf[t + s];
        buf[t] = (w == 0) ? fminf(a0, b0) : (w == 1) ? fmaxf(a0, b0) : (a0 + b0);
      }
      __syncthreads();
    }
    if (t == 0) gstat[w * NB + b] = buf[0];
    __syncthreads();
  }
}

// ---------------------------------------------------------------------------
// Final: min-max normalize + InstanceNorm + affine folded into one affine:
//   out = (L - mean_L) * rsqrt(var_L + 1e-5*rng^2) * gamma + beta   (rng > 0)
// ---------------------------------------------------------------------------
__global__ __launch_bounds__(256) void k_final(const float* __restrict__ gamma,
                                               const float* __restrict__ beta,
                                               const float* __restrict__ gstat,
                                               float* __restrict__ out) {
  const int idx = blockIdx.x * 256 + threadIdx.x;       // covers 16777216
  const int b   = idx >> 18;
  const float mn = gstat[0 * NB + b];
  const float mx = gstat[1 * NB + b];
  const float sm = gstat[2 * NB + b];
  const float sq = gstat[3 * NB + b];
  const float rng  = mx - mn;
  const float mean = sm * (1.0f / (float)IMG_ELEMS);
  const float var  = fmaxf(sq * (1.0f / (float)IMG_ELEMS) - mean * mean, 0.0f);
  const float L = out[idx];
  float res = (rng > 0.0f) ? (L - mean) * rsqrtf(var + 1e-5f * rng * rng) : 0.0f;
  out[idx] = res * gamma[0] + beta[0];
}

// ---------------------------------------------------------------------------
extern "C" void kernel_launch(void* const* d_in, const int* in_sizes, int n_in,
                              void* d_out, int out_size, void* d_ws, size_t ws_size,
                              hipStream_t stream) {
  (void)in_sizes; (void)n_in; (void)out_size; (void)ws_size;
  const float* x     = (const float*)d_in[0];
  const float* gamma = (const float*)d_in[1];
  const float* beta  = (const float*)d_in[2];
  float* out = (float*)d_out;

  float* wsf    = (float*)d_ws;
  float* FrT    = wsf;                       // 16777216
  float* FiT    = wsf + (size_t)PLANE;       // 16777216
  float* tab    = wsf + (size_t)2 * PLANE;   // 3584
  float* pmin   = tab + TAB_SZ;              // 1024
  float* pmax   = pmin + NPART;              // 1024
  float* psum   = pmax + NPART;              // 1024
  float* psumsq = psum + NPART;              // 1024
  float* gstat  = psumsq + NPART;            // 4*64

  const size_t smem1 = (size_t)(TAB_SZ + 2 * 32 * YPITCH) * sizeof(float);
  const size_t smem2 = smem1 + 1024 * sizeof(float);

  k_init  <<<7,    256,  0,     stream>>>(tab);
  k_phase1<<<1024, 1024, smem1, stream>>>(x, FrT, FiT, tab);
  k_phase2<<<1024, 1024, smem2, stream>>>(FrT, FiT, tab, out, pmin, pmax, psum, psumsq);
  k_reduce<<<NB,   16,   0,     stream>>>(pmin, pmax, psum, psumsq, gstat);
  k_final <<<65536, 256, 0,     stream>>>(gamma, beta, gstat, out);
}